// Loss_28183575396380
// MI455X (gfx1250) — compile-verified
//
#include <hip/hip_runtime.h>
#include <hip/hip_bf16.h>

typedef __attribute__((ext_vector_type(2))) float v2f;
typedef __attribute__((ext_vector_type(8))) float v8f;

#define BATCH        8
#define NPTS         4096
#define QT           4              // 16-row query tiles per wave
#define QPW          (QT * 16)      // 64 queries per wave
#define WAVES_PER_WG 8
#define CHUNK        1024           // staged target points per LDS chunk (16 KB as float4)

// Raw v_min_num_f32 / v_min3_num_f32: avoid clang's strict-FP canonicalize
// (v_max v,v,v) that otherwise doubles the min-path VALU work. Inputs are
// finite by construction, so IEEE NaN-quieting is unnecessary. Non-volatile
// so the scheduler may reorder them freely.
__device__ __forceinline__ float fmin_raw(float a, float b) {
  float r;
  asm("v_min_num_f32 %0, %1, %2" : "=v"(r) : "v"(a), "v"(b));
  return r;
}
__device__ __forceinline__ float fmin3_raw(float a, float b, float c) {
  float r;
  asm("v_min3_num_f32 %0, %1, %2, %3" : "=v"(r) : "v"(a), "v"(b), "v"(c));
  return r;
}

// ---------------------------------------------------------------------------
// NN kernel: for each query point q in set Q, compute
//     dmin[q] = min_n ( |t_n|^2 - 2 q . t_n )
// over all target points t (same batch).  The |q|^2 and clamp are applied in
// the reduction kernel (both commute with min).
// Core: V_WMMA_F32_16X16X4_F32 with A[m] = (-2qx,-2qy,-2qz,1),
//       B[n] = (tx,ty,tz,|t|^2)  -> D[m][n] = |t_n|^2 - 2 q_m . t_n.
// Two target tiles per inner iteration so the running min uses v_min3_num_f32
// (one VOP3 per element instead of two VOP2s).
// grid = (64, 2): x -> 8 workgroups per batch, y -> direction (0: pred->gt).
// ---------------------------------------------------------------------------
__global__ __launch_bounds__(256) void chamfer_nn_kernel(
    const float* __restrict__ pred, const float* __restrict__ gt,
    float* __restrict__ dmin /* [2][BATCH*NPTS] */) {
  __shared__ float4 s_t[CHUNK];

  const int dir = blockIdx.y;
  const float* qry = (dir == 0) ? pred : gt;
  const float* tgt = (dir == 0) ? gt : pred;
  float* out = dmin + (size_t)dir * (BATCH * NPTS);

  const int wave  = threadIdx.x >> 5;
  const int lane  = threadIdx.x & 31;
  const int gwave = blockIdx.x * WAVES_PER_WG + wave;   // 0..511
  const int batch = gwave >> 6;                         // 64 waves per batch
  const int qbase = (gwave & 63) * QPW;                 // query offset in batch

  const float* qb = qry + (size_t)batch * NPTS * 3;
  const float* tb = tgt + (size_t)batch * NPTS * 3;

  const int  mrow = lane & 15;
  const bool hiK  = (lane >= 16);

  // --- Build A operands (constant across the whole target sweep) ---
  v2f A[QT];
#pragma unroll
  for (int qt = 0; qt < QT; ++qt) {
    const float* qp = qb + 3 * (qbase + qt * 16 + mrow);
    const float qx = qp[0], qy = qp[1], qz = qp[2];
    A[qt].x = hiK ? (-2.0f * qz) : (-2.0f * qx);
    A[qt].y = hiK ? 1.0f         : (-2.0f * qy);
  }

  // --- Running-min accumulator tiles ---
  v8f minD[QT];
#pragma unroll
  for (int qt = 0; qt < QT; ++qt)
#pragma unroll
    for (int e = 0; e < 8; ++e) minD[qt][e] = 3.0e38f;

  const v8f czero = {};

  // --- Sweep all targets, staged through LDS in chunks ---
  for (int chunk = 0; chunk < NPTS; chunk += CHUNK) {
    // Cooperative stage: 256 threads x 4 points; precompute |t|^2 once per WG.
    for (int i = threadIdx.x; i < CHUNK; i += 256) {
      const float* tp = tb + 3 * (chunk + i);
      const float x = tp[0], y = tp[1], z = tp[2];
      float4 v; v.x = x; v.y = y; v.z = z; v.w = x * x + y * y + z * z;
      s_t[i] = v;
    }
    __syncthreads();

#pragma unroll 2
    for (int nt = 0; nt < CHUNK / 16; nt += 2) {
      // Two 16-point target tiles per iteration.
      const float4 t0 = s_t[nt * 16 + mrow];        // ds_load_b128, bank-clean
      const float4 t1 = s_t[nt * 16 + 16 + mrow];
      v2f B0, B1;
      B0.x = hiK ? t0.z : t0.x;
      B0.y = hiK ? t0.w : t0.y;
      B1.x = hiK ? t1.z : t1.x;
      B1.y = hiK ? t1.w : t1.y;
#pragma unroll
      for (int qt = 0; qt < QT; ++qt) {
        v8f d0 = __builtin_amdgcn_wmma_f32_16x16x4_f32(
            false, A[qt], false, B0, (short)0, czero, false, false);
        v8f d1 = __builtin_amdgcn_wmma_f32_16x16x4_f32(
            false, A[qt], false, B1, (short)0, czero, false, false);
#pragma unroll
        for (int e = 0; e < 8; ++e)
          minD[qt][e] = fmin3_raw(minD[qt][e], d0[e], d1[e]);
      }
    }
    __syncthreads();
  }

  // --- Horizontal min across the 16 lanes of each half (N dimension) ---
#pragma unroll
  for (int qt = 0; qt < QT; ++qt)
#pragma unroll
    for (int e = 0; e < 8; ++e) {
      float v = minD[qt][e];
      v = fmin_raw(v, __shfl_xor(v, 1));
      v = fmin_raw(v, __shfl_xor(v, 2));
      v = fmin_raw(v, __shfl_xor(v, 4));
      v = fmin_raw(v, __shfl_xor(v, 8));
      minD[qt][e] = v;
    }

  // Lane 0 holds rows r (e=0..7), lane 16 holds rows r+8.
  if ((lane & 15) == 0) {
    const int half = lane >> 4;
    float* op = out + (size_t)batch * NPTS + qbase;
#pragma unroll
    for (int qt = 0; qt < QT; ++qt)
#pragma unroll
      for (int e = 0; e < 8; ++e)
        op[qt * 16 + half * 8 + e] = minD[qt][e];
  }
}

// ---------------------------------------------------------------------------
// Epilogue: add |q|^2, clamp at 0, mean both directions, single scalar out.
// Single workgroup, deterministic accumulation order.
// ---------------------------------------------------------------------------
__global__ __launch_bounds__(256) void chamfer_reduce_kernel(
    const float* __restrict__ pred, const float* __restrict__ gt,
    const float* __restrict__ dmin, float* __restrict__ outv) {
  const int tid = threadIdx.x;
  const int total = BATCH * NPTS;  // 32768 queries per direction
  float acc = 0.0f;

  for (int i = tid; i < total; i += 256) {
    const float* qp = pred + 3 * (size_t)i;
    const float q2 = qp[0] * qp[0] + qp[1] * qp[1] + qp[2] * qp[2];
    acc += fmaxf(dmin[i] + q2, 0.0f);
  }
  for (int i = tid; i < total; i += 256) {
    const float* qp = gt + 3 * (size_t)i;
    const float q2 = qp[0] * qp[0] + qp[1] * qp[1] + qp[2] * qp[2];
    acc += fmaxf(dmin[total + i] + q2, 0.0f);
  }
  acc *= (1.0f / (float)total);  // both means have the same denominator

  __shared__ float red[256];
  red[tid] = acc;
  __syncthreads();
  for (int s = 128; s > 0; s >>= 1) {
    if (tid < s) red[tid] += red[tid + s];
    __syncthreads();
  }
  if (tid == 0) outv[0] = red[0];
}

extern "C" void kernel_launch(void* const* d_in, const int* in_sizes, int n_in,
                              void* d_out, int out_size, void* d_ws, size_t ws_size,
                              hipStream_t stream) {
  (void)in_sizes; (void)n_in; (void)out_size; (void)ws_size;
  const float* pred = (const float*)d_in[0];  // [8,4096,3] f32
  const float* gt   = (const float*)d_in[1];  // [8,4096,3] f32
  float* dmin = (float*)d_ws;                 // 2 * 8 * 4096 floats = 256 KB

  // 8 WGs per batch * 8 batches = 64; y = direction.
  chamfer_nn_kernel<<<dim3(8 * BATCH, 2), 256, 0, stream>>>(pred, gt, dmin);
  chamfer_reduce_kernel<<<1, 256, 0, stream>>>(pred, gt, dmin, (float*)d_out);
}